// VoxelRenderer_47141561041602
// MI455X (gfx1250) — compile-verified
//
#include <hip/hip_runtime.h>

// MI455X / gfx1250, wave32. Lane-per-ray NeRF voxel renderer.
// - direct gather from rgbsigma[b,y,x,z,c] layout (16B/voxel, global_load_b128)
// - 32-bit shift/add byte-offset addressing (strides are powers of two)
// - per-axis OOB masks folded into trilinear weights (6 checks, not 24)
// - online alpha compositing (no intermediate sample tensor)
// - ray rotation R @ dirs via V_WMMA_F32_16X16X4_F32 (shared-B matmul)
// - global_prefetch one sample ahead along the ray (constant voxel-space step)

typedef __attribute__((ext_vector_type(2))) float v2f;
typedef __attribute__((ext_vector_type(8))) float v8f;

#define GVX 128   // voxel grid resolution (Hv = Wv = Dv)
#define IMGW 128  // H = W
#define NSAMP 128

union V8u { v8f v; float f[8]; };

__global__ __launch_bounds__(256)
void voxel_render_kernel(const float* __restrict__ vox,    // [B,Hv,Wv,Dv,4]
                         const float* __restrict__ pose,   // [B,4,4]
                         const float* __restrict__ Kmat,   // [B,3,3]
                         const float* __restrict__ bbmin,  // [B,3]
                         const float* __restrict__ bbmax,  // [B,3]
                         float* __restrict__ out)          // [B,3,H,W]
{
    __shared__ float marshal[256 * 3];

    const int tid  = blockIdx.x * blockDim.x + threadIdx.x;
    const int lane = threadIdx.x & 31;
    float* wm = &marshal[((threadIdx.x >> 5) * 32) * 3];  // per-wave region

    const int b  = tid >> 14;          // / (128*128)  (wave never straddles batches)
    const int n  = tid & 16383;
    const int pi = n >> 7;             // image row  (H)
    const int pj = n & 127;            // image col  (W)

    // ---- camera-space direction -------------------------------------------
    const float fx = Kmat[b*9 + 0], fy = Kmat[b*9 + 4];
    const float cx = Kmat[b*9 + 2], cy = Kmat[b*9 + 5];
    const float dcx = ((float)pj - cx) / fx;
    const float dcy = -(((float)pi - cy) / fy);
    const float dcz = -1.0f;

    // ---- rotate by pose[:3,:3] via V_WMMA_F32_16X16X4_F32 -----------------
    // D(16 rays x 3) = A(16 rays x 4 = [dx,dy,dz,0]) * B(4x16 = R^T padded)
    wm[lane*3 + 0] = dcx;
    wm[lane*3 + 1] = dcy;
    wm[lane*3 + 2] = dcz;
    __syncthreads();

    const int r = lane & 15;
    v2f A0, A1, Bm;
    // A layout (f32 16x4): VGPR0 lanes0-15 = K0, lanes16-31 = K2;
    //                      VGPR1 lanes0-15 = K1, lanes16-31 = K3.
    A0.x = (lane < 16) ? wm[r*3 + 0]        : wm[r*3 + 2];
    A0.y = (lane < 16) ? wm[r*3 + 1]        : 0.0f;
    A1.x = (lane < 16) ? wm[(r+16)*3 + 0]   : wm[(r+16)*3 + 2];
    A1.y = (lane < 16) ? wm[(r+16)*3 + 1]   : 0.0f;
    // B[k][n] = R[n][k] = pose[b][n][k], zero-padded
    {
        const int k0 = (lane < 16) ? 0 : 2;
        Bm.x = (r < 3) ? pose[b*16 + r*4 + k0] : 0.0f;
        Bm.y = (lane < 16 && r < 3) ? pose[b*16 + r*4 + 1] : 0.0f;
    }
    __syncthreads();

    v8f cz = {0.f,0.f,0.f,0.f,0.f,0.f,0.f,0.f};
    V8u d0, d1;
    d0.v = __builtin_amdgcn_wmma_f32_16x16x4_f32(false, A0, false, Bm,
                                                 (short)0, cz, false, false);
    d1.v = __builtin_amdgcn_wmma_f32_16x16x4_f32(false, A1, false, Bm,
                                                 (short)0, cz, false, false);

    // scatter D (rows = local rays, cols 0..2 = world-dir components) to LDS
    if (r < 3) {
        #pragma unroll
        for (int q = 0; q < 8; ++q) {
            const int row = (lane < 16) ? q : (8 + q);
            wm[row*3 + r]      = d0.f[q];   // tile0 -> local rays 0..15
            wm[(16+row)*3 + r] = d1.f[q];   // tile1 -> local rays 16..31
        }
    }
    __syncthreads();

    float dwx = wm[lane*3 + 0];
    float dwy = wm[lane*3 + 1];
    float dwz = wm[lane*3 + 2];

    // ---- normalize, slab-intersect bbox -----------------------------------
    float nn = sqrtf(dwx*dwx + dwy*dwy + dwz*dwz);
    nn = fmaxf(nn, 1e-12f);
    dwx /= nn; dwy /= nn; dwz /= nn;

    const float oxw = pose[b*16 + 3], oyw = pose[b*16 + 7], ozw = pose[b*16 + 11];
    const float bmnx = bbmin[b*3+0], bmny = bbmin[b*3+1], bmnz = bbmin[b*3+2];
    const float bmxx = bbmax[b*3+0], bmxy = bbmax[b*3+1], bmxz = bbmax[b*3+2];

    const float ivx = 1.0f / (dwx + 1e-10f);
    const float ivy = 1.0f / (dwy + 1e-10f);
    const float ivz = 1.0f / (dwz + 1e-10f);
    const float tax = (bmnx - oxw) * ivx, tbx = (bmxx - oxw) * ivx;
    const float tay = (bmny - oyw) * ivy, tby = (bmxy - oyw) * ivy;
    const float taz = (bmnz - ozw) * ivz, tbz = (bmxz - ozw) * ivz;
    const float lox = fminf(tax, tbx), hix = fmaxf(tax, tbx);
    const float loy = fminf(tay, tby), hiy = fmaxf(tay, tby);
    const float loz = fminf(taz, tbz), hiz = fmaxf(taz, tbz);
    const float nearT = fmaxf(fmaxf(lox, fmaxf(loy, loz)), 0.1f);
    const float farT  = fmaxf(fminf(hix, fminf(hiy, hiz)), nearT + 0.1f);
    const float dt    = (farT - nearT) * (1.0f / 127.0f);

    // voxel-space affine:  v(t) = m*t + b0   (align_corners=True)
    const float ssx = 127.0f / (bmxx - bmnx + 1e-10f);
    const float ssy = 127.0f / (bmxy - bmny + 1e-10f);
    const float ssz = 127.0f / (bmxz - bmnz + 1e-10f);
    const float mx = dwx * ssx, my = dwy * ssy, mz = dwz * ssz;
    const float bx0 = (oxw - bmnx) * ssx;
    const float by0 = (oyw - bmny) * ssy;
    const float bz0 = (ozw - bmnz) * ssz;
    const float stx = mx * dt, sty = my * dt, stz = mz * dt;   // voxel step / sample

    const char* vbc = (const char*)(vox + (size_t)b * ((size_t)GVX * GVX * GVX * 4));

    float T = 1.0f, accW = 0.0f, accR = 0.0f, accG = 0.0f, accB = 0.0f;

    #pragma unroll 2
    for (int s = 0; s < NSAMP; ++s) {
        const float t = fmaf(dt, (float)s, nearT);
        const float x = fmaf(mx, t, bx0);        // -> Wv axis
        const float y = fmaf(my, t, by0);        // -> Hv axis
        const float z = fmaf(mz, t, bz0);        // -> Dv axis

        const float xf = floorf(x), yf = floorf(y), zf = floorf(z);
        const float xd = x - xf, yd = y - yf, zd = z - zf;
        const int ix = (int)xf, iy = (int)yf, iz = (int)zf;

        // clamped per-axis indices (clamp BEFORE +1: no signed overflow)
        const int xc0 = min(max(ix, 0), 127), xc1 = min(max(ix, -1), 126) + 1;
        const int yc0 = min(max(iy, 0), 127), yc1 = min(max(iy, -1), 126) + 1;
        const int zc0 = min(max(iz, 0), 127), zc1 = min(max(iz, -1), 126) + 1;

        // byte offsets: voxel (y,x,z) -> (y<<18) + (x<<11) + (z<<4)
        const int a00 = (yc0 << 18) + (xc0 << 11);   // (y0,x0)
        const int a01 = (yc0 << 18) + (xc1 << 11);   // (y0,x1)
        const int a10 = (yc1 << 18) + (xc0 << 11);   // (y1,x0)
        const int a11 = (yc1 << 18) + (xc1 << 11);   // (y1,x1)
        const int o0 = zc0 << 4, o1 = zc1 << 4;

        // issue all 8 gathers up front (independent b128 loads)
        const float4 v000 = *(const float4*)(vbc + (a00 + o0));
        const float4 v001 = *(const float4*)(vbc + (a01 + o0));
        const float4 v010 = *(const float4*)(vbc + (a10 + o0));
        const float4 v011 = *(const float4*)(vbc + (a11 + o0));
        const float4 v100 = *(const float4*)(vbc + (a00 + o1));
        const float4 v101 = *(const float4*)(vbc + (a01 + o1));
        const float4 v110 = *(const float4*)(vbc + (a10 + o1));
        const float4 v111 = *(const float4*)(vbc + (a11 + o1));

        // prefetch next sample (constant voxel-space step; address is a hint)
        {
            const int pxn = min(max((int)(x + stx), 0), 127);
            const int pyn = min(max((int)(y + sty), 0), 127);
            const int pzn = min(max((int)(z + stz), 0), 127);
            __builtin_prefetch(vbc + ((pyn << 18) + (pxn << 11) + (pzn << 4)), 0, 1);
        }

        // per-axis weights with OOB mask folded in (padding_mode=zeros)
        const float wx0 = (ix >= 0  && ix < 128) ? (1.0f - xd) : 0.0f;
        const float wx1 = (ix >= -1 && ix < 127) ? xd          : 0.0f;
        const float wy0 = (iy >= 0  && iy < 128) ? (1.0f - yd) : 0.0f;
        const float wy1 = (iy >= -1 && iy < 127) ? yd          : 0.0f;
        const float wz0 = (iz >= 0  && iz < 128) ? (1.0f - zd) : 0.0f;
        const float wz1 = (iz >= -1 && iz < 127) ? zd          : 0.0f;

        const float w00 = wz0 * wy0, w01 = wz0 * wy1;   // (z, y)
        const float w10 = wz1 * wy0, w11 = wz1 * wy1;

        float sR = 0.f, sG = 0.f, sB = 0.f, sS = 0.f;
        #define TAP(V, W) do { const float _w = (W);              \
            sR = fmaf(_w, (V).x, sR); sG = fmaf(_w, (V).y, sG);   \
            sB = fmaf(_w, (V).z, sB); sS = fmaf(_w, (V).w, sS); } while (0)
        TAP(v000, w00 * wx0);  TAP(v001, w00 * wx1);
        TAP(v010, w01 * wx0);  TAP(v011, w01 * wx1);
        TAP(v100, w10 * wx0);  TAP(v101, w10 * wx1);
        TAP(v110, w11 * wx0);  TAP(v111, w11 * wx1);
        #undef TAP

        const float dist  = (s == NSAMP - 1) ? 1e10f : dt;
        const float alpha = 1.0f - __expf(-fmaxf(sS, 0.0f) * dist);
        const float w_s   = alpha * T;
        accR = fmaf(w_s, sR, accR);
        accG = fmaf(w_s, sG, accG);
        accB = fmaf(w_s, sB, accB);
        accW += w_s;
        T *= (1.0f - alpha + 1e-10f);
    }

    const float bg = 1.0f - accW;
    const int obase = ((b * 3 + 0) * IMGW + pi) * IMGW + pj;
    out[obase]                   = accR + bg;
    out[obase + IMGW * IMGW]     = accG + bg;
    out[obase + 2 * IMGW * IMGW] = accB + bg;
}

extern "C" void kernel_launch(void* const* d_in, const int* in_sizes, int n_in,
                              void* d_out, int out_size, void* d_ws, size_t ws_size,
                              hipStream_t stream) {
    const float* vox  = (const float*)d_in[0];
    const float* pose = (const float*)d_in[1];
    const float* K    = (const float*)d_in[2];
    const float* bmn  = (const float*)d_in[3];
    const float* bmx  = (const float*)d_in[4];
    float* out = (float*)d_out;

    const int B = in_sizes[1] / 16;          // pose is [B,4,4]
    const int total = B * IMGW * IMGW;       // one thread per ray
    voxel_render_kernel<<<total / 256, 256, 0, stream>>>(vox, pose, K, bmn, bmx, out);
}